// GraphSAGE_65240553226635
// MI455X (gfx1250) — compile-verified
//
#include <hip/hip_runtime.h>

typedef __attribute__((ext_vector_type(2))) float v2f;
typedef __attribute__((ext_vector_type(8))) float v8f;

static constexpr int KDIM = 128;   // D_IN == D_H == 128 (K for both layers)

// ---------------- utility kernels ----------------

__global__ void zero_kernel(float* __restrict__ p, long long n) {
  long long i = (long long)blockIdx.x * blockDim.x + threadIdx.x;
  long long stride = (long long)gridDim.x * blockDim.x;
  for (; i < n; i += stride) p[i] = 0.0f;
}

__global__ void degree_kernel(const int* __restrict__ dst, float* __restrict__ deg,
                              int E) {
  int e = blockIdx.x * blockDim.x + threadIdx.x;
  if (e < E) atomicAdd(deg + dst[e], 1.0f);
}

__global__ void invdeg_kernel(float* __restrict__ deg, int n) {
  int i = blockIdx.x * blockDim.x + threadIdx.x;
  if (i < n) {
    float d = deg[i];
    deg[i] = (d > 0.0f) ? (1.0f / d) : 0.0f;   // deg integral: 1/max(deg,1) == 1/deg
  }
}

// One wave32 per edge; each lane moves 4 contiguous features (4*32 = 128 = KDIM).
// Row read is a coalesced 512B burst; atomics land on a contiguous dst row in L2.
__global__ void scatter_kernel(const float* __restrict__ feat,
                               const int* __restrict__ src,
                               const int* __restrict__ dst,
                               float* __restrict__ msg, int E) {
  int tid  = blockIdx.x * blockDim.x + threadIdx.x;
  int e    = tid >> 5;
  int lane = tid & 31;
  if (e >= E) return;
  int s = src[e];
  int d = dst[e];
  const float4* row = (const float4*)(feat + (size_t)s * KDIM);
  float4 v = row[lane];
  float* m = msg + (size_t)d * KDIM + lane * 4;
  atomicAdd(m + 0, v.x);
  atomicAdd(m + 1, v.y);
  atomicAdd(m + 2, v.z);
  atomicAdd(m + 3, v.w);
}

// ---------------- fused SAGE layer: out = act((msg*invdeg)@Wl^T + b + root@Wr^T) ---
//
// One wave computes a 16x16 output tile with chained V_WMMA_F32_16X16X4_F32.
// A 16x4 f32 layout (ISA 7.12.2): lane m=lane&15 holds row m; VGPR0/1 hold
// K = 4s+2*hi .. +1 where hi = lane>>4.  B 4x16 mirrors this with N striped
// across lanes, so the B operand (W^T) is a float2 from weight row (col_base+m).
// D layout: VGPR r -> M = 8*hi + r, N = lane&15.
template <int NCOLS, bool RELU>
__global__ void sage_gemm_kernel(const float* __restrict__ msg,     // [n,K] raw sums
                                 const float* __restrict__ root,    // [n,K]
                                 const float* __restrict__ invdeg,  // [n]
                                 const float* __restrict__ Wl,      // [NCOLS,K]
                                 const float* __restrict__ bias,    // [NCOLS]
                                 const float* __restrict__ Wr,      // [NCOLS,K]
                                 float* __restrict__ out) {         // [n,NCOLS]
  const int lane     = threadIdx.x & 31;
  const int wave     = threadIdx.x >> 5;
  const int row_base = blockIdx.x * 16;
  const int col_base = wave * 16;          // blockDim.x = 32 * NCOLS/16
  const int m  = lane & 15;
  const int hi = lane >> 4;

  const int arow = row_base + m;           // A row owned by this lane
  const int wrow = col_base + m;           // weight row == output column

  const float scale = invdeg[arow];
  const float2* a1 = (const float2*)(msg  + (size_t)arow * KDIM + 2 * hi);
  const float2* a2 = (const float2*)(root + (size_t)arow * KDIM + 2 * hi);
  const float2* bl = (const float2*)(Wl   + (size_t)wrow * KDIM + 2 * hi);
  const float2* br = (const float2*)(Wr   + (size_t)wrow * KDIM + 2 * hi);

  v8f acc = {};
#pragma unroll
  for (int s = 0; s < KDIM / 4; ++s) {     // (msg * invdeg) @ Wl^T
    float2 af = a1[2 * s];                 // K = 4s + 2*hi .. +1
    float2 bf = bl[2 * s];
    v2f a = {af.x * scale, af.y * scale};
    v2f b = {bf.x, bf.y};
    acc = __builtin_amdgcn_wmma_f32_16x16x4_f32(false, a, false, b, (short)0, acc,
                                                false, false);
  }
#pragma unroll
  for (int s = 0; s < KDIM / 4; ++s) {     // + root @ Wr^T
    float2 af = a2[2 * s];
    float2 bf = br[2 * s];
    v2f a = {af.x, af.y};
    v2f b = {bf.x, bf.y};
    acc = __builtin_amdgcn_wmma_f32_16x16x4_f32(false, a, false, b, (short)0, acc,
                                                false, false);
  }

  const float bv = bias[wrow];
  float* op = out + (size_t)(row_base + 8 * hi) * NCOLS + wrow;
#pragma unroll
  for (int r = 0; r < 8; ++r) {
    float v = acc[r] + bv;
    if (RELU) v = fmaxf(v, 0.0f);
    op[(size_t)r * NCOLS] = v;
  }
}

// ---------------- driver ----------------

extern "C" void kernel_launch(void* const* d_in, const int* in_sizes, int n_in,
                              void* d_out, int out_size, void* d_ws, size_t ws_size,
                              hipStream_t stream) {
  const float* x   = (const float*)d_in[0];
  const int*   ei  = (const int*)d_in[1];     // [2,E] row-major: src then dst
  const float* W1l = (const float*)d_in[2];
  const float* b1  = (const float*)d_in[3];
  const float* W1r = (const float*)d_in[4];
  const float* W2l = (const float*)d_in[5];
  const float* b2  = (const float*)d_in[6];
  const float* W2r = (const float*)d_in[7];
  float* out = (float*)d_out;

  const int n = in_sizes[0] / KDIM;   // 50000 (divisible by 16)
  const int E = in_sizes[1] / 2;      // 800000
  const int* src = ei;
  const int* dst = ei + E;

  // workspace: deg[n] | msg[n*128] | h[n*128]  (~51 MB)
  float* deg = (float*)d_ws;
  float* msg = deg + n;
  float* h   = msg + (size_t)n * KDIM;

  const unsigned scat_blocks = (unsigned)(((long long)E * 32 + 255) / 256);

  // layer 1
  zero_kernel<<<2048, 256, 0, stream>>>(deg, (long long)n * (KDIM + 1));
  degree_kernel<<<(E + 255) / 256, 256, 0, stream>>>(dst, deg, E);
  invdeg_kernel<<<(n + 255) / 256, 256, 0, stream>>>(deg, n);
  scatter_kernel<<<scat_blocks, 256, 0, stream>>>(x, src, dst, msg, E);
  sage_gemm_kernel<128, true><<<n / 16, 256, 0, stream>>>(msg, x, deg, W1l, b1, W1r, h);

  // layer 2 (reuse msg buffer)
  zero_kernel<<<2048, 256, 0, stream>>>(msg, (long long)n * KDIM);
  scatter_kernel<<<scat_blocks, 256, 0, stream>>>(h, src, dst, msg, E);
  sage_gemm_kernel<64, false><<<n / 16, 128, 0, stream>>>(msg, h, deg, W2l, b2, W2r, out);
}